// Resample_70832600645887
// MI455X (gfx1250) — compile-verified
//
#include <hip/hip_runtime.h>

// fp32 WMMA fragment types (wave32): A/B = 16x4 / 4x16 f32 -> 2 VGPRs; C/D 16x16 f32 -> 8 VGPRs
typedef float v2f __attribute__((ext_vector_type(2)));
typedef float v8f __attribute__((ext_vector_type(8)));

#define NN          8388608      // input length
#define FF          1023         // filter taps
#define P_PER_PHASE 4194304      // outputs per polyphase branch (out_len = 3 * this)
#define PB          1024         // outputs (per phase) handled by one workgroup
#define BLOCK       256          // 8 waves of 32
#define NWAVES      (BLOCK / 32)
#define NGROUP      6            // 6 * 64 = 384 padded taps >= ceil(1023/3)=341
#define XW          (2 * PB + 448)   // staged x window (floats); covers all tile reads
#define GUARD       48               // accumulator guard band (tile diagonal spill)
#define ACCSZ       (PB + 2 * GUARD)

__global__ __launch_bounds__(BLOCK)
void resample_poly_wmma(const float* __restrict__ x,
                        const float* __restrict__ filt,
                        float* __restrict__ out) {
    __shared__ float acc[ACCSZ];  // guarded output accumulator: index = (p - P0) + GUARD
    __shared__ float xw[XW];      // staged input window (zero-padded at signal edges)

    const int tid = threadIdx.x;

    // Output slot o = i mod 3 maps to polyphase residue r and base input index Q:
    //   o=0: r=1, q=170+2p ; o=1: r=0, q=171+2p ; o=2: r=2, q=171+2p
    const int o = blockIdx.y;
    const int r = (o == 0) ? 1 : (o == 1) ? 0 : 2;
    const int Q = (o == 0) ? 170 : 171;

    const int P0 = blockIdx.x * PB;      // this block covers phase outputs [P0, P0+PB)
    const int W0 = Q + 2 * P0 - 416;     // global x index of xw[0]

    // ---- zero the accumulator (guard bands soak up halo-tile spill) ----
    for (int i = tid; i < ACCSZ; i += BLOCK) acc[i] = 0.0f;

    // ---- stage x window into LDS: coalesced, branchless edge handling ----
    for (int i = tid; i < XW; i += BLOCK) {
        const unsigned ui = (unsigned)(W0 + i);             // negative -> huge unsigned
        const unsigned cl = (ui < (unsigned)NN) ? ui : 0u;  // cndmask on address
        const float    v  = x[cl];                          // unconditional coalesced load
        xw[i] = (ui < (unsigned)NN) ? v : 0.0f;             // cndmask on data
    }
    __syncthreads();

    const int lane = tid & 31;
    const int wave = tid >> 5;
    const int row  = lane & 15;   // A: M index / B,D: N index (same lane field)
    const int kh   = lane >> 4;   // K-half select (lanes 16-31 carry K=2,3)

    // B fragments (per-phase constants). K order permuted so A reads ascending memory:
    //   A[m][2kh]   = x[c + 2m - (2kh+1)]  <->  B[2kh][n]   = hr[64g + 4n + 2kh + 1]
    //   A[m][2kh+1] = x[c + 2m -  2kh  ]  <->  B[2kh+1][n] = hr[64g + 4n + 2kh    ]
    // (same k' pairs both sides, so D is unchanged; hr[t] = h[3t + r], 0 for t >= 341)
    v2f bfrag[NGROUP];
#pragma unroll
    for (int g = 0; g < NGROUP; ++g) {
        const int t0 = 64 * g + 4 * row + 2 * kh;
        const int j0 = 3 * t0 + r;                 // h index of tap t0
        const int j1 = j0 + 3;                     // h index of tap t0+1
        bfrag[g].x = (j1 < FF) ? filt[j1] : 0.0f;  // tap t0+1 pairs with xw[i0-1]
        bfrag[g].y = (j0 < FF) ? filt[j0] : 0.0f;  // tap t0   pairs with xw[i0]
    }

    // Tiles tau whose output span [16*tau, 16*tau+45] intersects [P0, P0+PB)
    const int tlo = (P0 - 30) >> 4;            // arithmetic shift = floor div
    const int thi = (P0 + PB - 1) >> 4;

    for (int tau = tlo + wave; tau <= thi; tau += NWAVES) {
        const int p0 = tau << 4;
        // LDS index of the K=2kh+1 element (always in range [4, XW-4)):
        //   i0 = 2*(p0 - P0) + 2*row - 2*kh + 416 - 64*g ; pair is {xw[i0-1], xw[i0]}
        const int ibase = 2 * (p0 - P0) + 2 * row - 2 * kh + 416;
        v8f c = {0.f, 0.f, 0.f, 0.f, 0.f, 0.f, 0.f, 0.f};
#pragma unroll
        for (int g = 0; g < NGROUP; ++g) {
            const int i0 = ibase - 64 * g;
            v2f a;
            a.x = xw[i0 - 1];   // K = 2kh   (ascending memory order -> no reg swaps)
            a.y = xw[i0];       // K = 2kh+1
            c = __builtin_amdgcn_wmma_f32_16x16x4_f32(
                    /*neg_a=*/false, a, /*neg_b=*/false, bfrag[g],
                    /*c_mod=*/(short)0, c, /*reuse_a=*/false, /*reuse_b=*/false);
        }
        // Cell (m = e + 8*kh, n = row) -> output p = p0 + m + 2n.
        // Guarded accumulator makes all 8 scatter adds unconditional (no exec masking).
        const int sbase = p0 - P0 + 8 * kh + 2 * row + GUARD;
#pragma unroll
        for (int e = 0; e < 8; ++e) {
            atomicAdd(&acc[sbase + e], c[e]);  // ds_add_f32, consecutive offsets
        }
    }
    __syncthreads();

    // out[3p + o] = 3 * sum  (x_up carries the 'up' gain in the reference)
    for (int s = tid; s < PB; s += BLOCK) {
        const int p = P0 + s;
        out[3 * p + o] = 3.0f * acc[s + GUARD];
    }
}

extern "C" void kernel_launch(void* const* d_in, const int* in_sizes, int n_in,
                              void* d_out, int out_size, void* d_ws, size_t ws_size,
                              hipStream_t stream) {
    // setup_inputs order: x (f32, N), up (int,1), down (int,1), filter_coeffs (f32, F)
    const float* x    = (const float*)d_in[0];
    const float* filt = (const float*)d_in[3];
    float*       out  = (float*)d_out;
    (void)in_sizes; (void)n_in; (void)d_ws; (void)ws_size; (void)out_size;

    dim3 grid(P_PER_PHASE / PB, 3, 1);   // 4096 blocks x 3 polyphase branches
    dim3 block(BLOCK, 1, 1);
    resample_poly_wmma<<<grid, block, 0, stream>>>(x, filt, out);
}